// SpatialGraphConv_91311004713311
// MI455X (gfx1250) — compile-verified
//
#include <hip/hip_runtime.h>

// Problem constants (from the reference)
#define N_NODES 100000
#define CIN     32
#define COUT    32
#define TSTEPS  12
#define NEDGE   1600000
#define DT      (COUT * TSTEPS)   // 384 floats per node row of h / out

typedef __attribute__((ext_vector_type(2))) float v2f;
typedef __attribute__((ext_vector_type(8))) float v8f;

// ---------------------------------------------------------------- degree ----
__global__ __launch_bounds__(256) void init_deg_kernel(float* __restrict__ deg) {
    int n = blockIdx.x * blockDim.x + threadIdx.x;
    if (n < N_NODES) deg[n] = 1.0f;   // self-loop contributes weight 1
}

__global__ __launch_bounds__(256) void accum_deg_kernel(const long long* __restrict__ ei,
                                                        const float* __restrict__ ea,
                                                        float* __restrict__ deg) {
    int e = blockIdx.x * blockDim.x + threadIdx.x;
    if (e < NEDGE) {
        int d = (int)ei[NEDGE + e];           // dst row of edge_index
        atomicAdd(&deg[d], ea[e]);            // global_atomic_add_f32 (no return)
    }
}

__global__ __launch_bounds__(256) void dinv_kernel(float* __restrict__ deg) {
    int n = blockIdx.x * blockDim.x + threadIdx.x;
    if (n < N_NODES) {
        float v = deg[n];
        deg[n] = (v > 0.0f) ? rsqrtf(v) : 0.0f;   // in place: deg -> dinv
    }
}

// ------------------------------------------------------------- WMMA GEMM ----
// h[n, d, t] = sum_c x[n, c, t] * W[c, d]
// GEMM view: rows m = n*T + t (M = 1.2M, multiple of 16), cols = d (32), K = c (32).
// One wave per 16-row m-tile; two f32 accumulators cover d=0..15 and d=16..31.
// Uses V_WMMA_F32_16X16X4_F32 (fp32 matrix pipe, exact numerics), K-loop of 8.
__global__ __launch_bounds__(256) void gemm_wmma_kernel(const float* __restrict__ x,
                                                        const float* __restrict__ W,
                                                        float* __restrict__ h) {
    __shared__ float Wl[CIN * COUT];          // 4 KB of LDS, B operand source
    const int tid = threadIdx.x;
    for (int i = tid; i < CIN * COUT; i += 256) Wl[i] = W[i];
    __syncthreads();

    const int lane = tid & 31;
    const int wave = tid >> 5;
    const int m_base = (blockIdx.x * 8 + wave) * 16;   // exact: N*T/16 tiles

    // A-operand addressing (ISA 16x4 f32 layout):
    //   lanes 0..15 hold M=lane,     K = kk, kk+1
    //   lanes 16..31 hold M=lane-16, K = kk+2, kk+3
    const int mrow  = m_base + (lane & 15);
    const int node  = mrow / TSTEPS;
    const int tt    = mrow % TSTEPS;
    const size_t xbase = (size_t)node * (CIN * TSTEPS) + tt;  // x[n, c, t]: + c*T
    const int khalf = (lane >> 4) << 1;
    const int dcol  = lane & 15;

    v8f c0 = {};   // d = 0..15
    v8f c1 = {};   // d = 16..31
#pragma unroll
    for (int kk = 0; kk < CIN; kk += 4) {
        const int k0 = kk + khalf;
        v2f a;
        a.x = x[xbase + (size_t)k0 * TSTEPS];
        a.y = x[xbase + (size_t)(k0 + 1) * TSTEPS];
        // B operand (4x16, K rows striped like A, N = lane&15) from LDS
        v2f b0, b1;
        b0.x = Wl[k0 * COUT + dcol];
        b0.y = Wl[(k0 + 1) * COUT + dcol];
        b1.x = Wl[k0 * COUT + 16 + dcol];
        b1.y = Wl[(k0 + 1) * COUT + 16 + dcol];
        c0 = __builtin_amdgcn_wmma_f32_16x16x4_f32(false, a, false, b0, (short)0, c0, false, false);
        c1 = __builtin_amdgcn_wmma_f32_16x16x4_f32(false, a, false, b1, (short)0, c1, false, false);
    }

    // D layout: VGPR r -> M = r (lanes 0..15) / r+8 (lanes 16..31); N = lane&15
    const int mhalf = (lane >> 4) << 3;
#pragma unroll
    for (int r = 0; r < 8; ++r) {
        const int mg = m_base + mhalf + r;
        const int nd = mg / TSTEPS;
        const int t2 = mg % TSTEPS;
        const size_t o = (size_t)nd * DT + t2;
        h[o + (size_t)dcol * TSTEPS]        = c0[r];
        h[o + (size_t)(dcol + 16) * TSTEPS] = c1[r];
    }
}

// -------------------------------------------------- self-loop + bias init ---
// out[n, d, t] = b[d] + dinv[n]^2 * h[n, d, t]   (must run before edge scatter)
__global__ __launch_bounds__(256) void selfbias_kernel(const float* __restrict__ h,
                                                       const float* __restrict__ dinv,
                                                       const float* __restrict__ bias,
                                                       float* __restrict__ out) {
    int i = blockIdx.x * blockDim.x + threadIdx.x;   // < N*384 = 38.4M
    if (i < N_NODES * DT) {
        int n = i / DT;
        int d = (i % DT) / TSTEPS;
        float s = dinv[n];
        out[i] = bias[d] + s * s * h[i];
    }
}

// ------------------------------------------------------------ edge scatter --
// One wave32 per edge: out[dst,:,:] += coef * h[src,:,:]  (384 floats,
// 12 fully coalesced 128B atomic bursts). h is L2-resident (154MB < 192MB L2),
// so the gather hits L2; atomics are non-returning (STOREcnt path).
__global__ __launch_bounds__(256) void scatter_kernel(const long long* __restrict__ ei,
                                                      const float* __restrict__ ea,
                                                      const float* __restrict__ dinv,
                                                      const float* __restrict__ h,
                                                      float* __restrict__ out) {
    const int wid  = blockIdx.x * 8 + (threadIdx.x >> 5);   // edge id (exact grid)
    const int lane = threadIdx.x & 31;
    const int s = (int)ei[wid];
    const int d = (int)ei[NEDGE + wid];
    const float coef = dinv[s] * ea[wid] * dinv[d];
    const float* __restrict__ hs = h + (size_t)s * DT;
    float* __restrict__ od = out + (size_t)d * DT;
    __builtin_prefetch(hs, 0, 1);                 // global_prefetch_b8 (gfx1250)
#pragma unroll
    for (int j = 0; j < DT / 32; ++j) {
        atomicAdd(&od[j * 32 + lane], coef * hs[j * 32 + lane]);
    }
}

// ----------------------------------------------------------------- driver ---
extern "C" void kernel_launch(void* const* d_in, const int* in_sizes, int n_in,
                              void* d_out, int out_size, void* d_ws, size_t ws_size,
                              hipStream_t stream) {
    const float*     x  = (const float*)d_in[0];      // [N, CIN, T]
    const long long* ei = (const long long*)d_in[1];  // [2, E] int64
    const float*     ea = (const float*)d_in[2];      // [E]
    const float*     W  = (const float*)d_in[3];      // [CIN, COUT]
    const float*     b  = (const float*)d_in[4];      // [COUT]
    float* out = (float*)d_out;                       // [N, COUT, T]

    // Workspace: [dinv: N floats (padded)] [h: N*384 floats ~ 154MB, L2-resident]
    float* deg = (float*)d_ws;
    float* h   = deg + 102400;                        // 102400 = N padded to 256

    // 1) degree (with self loops) -> dinv
    init_deg_kernel<<<(N_NODES + 255) / 256, 256, 0, stream>>>(deg);
    accum_deg_kernel<<<(NEDGE + 255) / 256, 256, 0, stream>>>(ei, ea, deg);
    dinv_kernel<<<(N_NODES + 255) / 256, 256, 0, stream>>>(deg);

    // 2) h = einsum('nct,cd->ndt') via fp32 WMMA. N*T/16 = 75000 tiles, 8 waves/block.
    gemm_wmma_kernel<<<(N_NODES * TSTEPS / 16) / 8, 256, 0, stream>>>(x, W, h);

    // 3) out = b + dinv^2 * h   (self-loop term, initializes out)
    selfbias_kernel<<<(N_NODES * DT + 255) / 256, 256, 0, stream>>>(h, deg, b, out);

    // 4) edge scatter: one wave per edge, E/8 = 200000 blocks (exact)
    scatter_kernel<<<NEDGE / 8, 256, 0, stream>>>(ei, ea, deg, h, out);
}